// TripleMGQEEmbedding_45930380264187
// MI455X (gfx1250) — compile-verified
//
#include <hip/hip_runtime.h>

typedef __attribute__((ext_vector_type(2))) float v2f;
typedef __attribute__((ext_vector_type(8))) float v8f;

#define VOCAB    1000000
#define EMB      128
#define D        8
#define SUB      16
#define KC       256
#define NTOK     65536
#define CUT_HEAD 990000
#define CUT_MID  900000
#define BN_EPS   0.001f

#define WAVES  8
#define TPB    (WAVES * 32)
#define TILES  (NTOK / 16)       // 4096
#define BLOCKS (TILES / WAVES)   // 512

// workspace layout (float offsets)
#define OFF_SUM   0      // [3][256]
#define OFF_SUM2  768    // [3][256]
#define OFF_CNT   1536   // [3] (+1 pad)
#define OFF_MEAN  1540   // [3][256]
#define OFF_ISTD  2308   // [3][256]
#define WS_FLOATS 3076

__device__ __forceinline__ int part_of(int id) {
    return (id >= CUT_HEAD) ? 0 : ((id >= CUT_MID) ? 1 : 2);
}
__device__ __forceinline__ int kp_of(int p) {
    return (p == 0) ? 256 : ((p == 1) ? 128 : 32);
}

__global__ void __launch_bounds__(256) init_ws_kernel(float* ws) {
    for (int i = threadIdx.x; i < WS_FLOATS; i += 256) ws[i] = 0.0f;
}

// ---------------- Pass 1: per-(partition,k) sums of resp and resp^2 ----------------
__global__ void __launch_bounds__(TPB) mgqe_stats_kernel(
    const float* __restrict__ emb, const float* __restrict__ cent,
    const int* __restrict__ ids, float* __restrict__ ws)
{
    __shared__ float cn_lds[D][KC];
    __shared__ float s_sum[3][KC];
    __shared__ float s_sum2[3][KC];
    __shared__ float s_cnt[4];
    __shared__ float xn_lds[WAVES][16][D];
    __shared__ int   pt_lds[WAVES][16];

    const int tid = threadIdx.x;
    if (tid < 3) s_cnt[tid] = 0.0f;
    for (int i = tid; i < 3 * KC; i += TPB) { (&s_sum[0][0])[i] = 0.0f; (&s_sum2[0][0])[i] = 0.0f; }
    for (int i = tid; i < D * KC; i += TPB) {
        const int d = i >> 8, k = i & 255;
        const float* c = cent + (size_t)(d * KC + k) * SUB;
        float s = 0.0f;
        #pragma unroll
        for (int j = 0; j < SUB; j++) { const float cv = c[j]; s += cv * cv; }
        cn_lds[d][k] = s;
    }
    __syncthreads();

    const int wave = tid >> 5, lane = tid & 31;
    const int col = lane & 15, half = lane >> 4;
    const int tile = blockIdx.x * WAVES + wave;
    const int tok0 = tile * 16;

    const int myid = ids[tok0 + col];
    const int myp  = part_of(myid);
    if (half == 0) { pt_lds[wave][col] = myp; atomicAdd(&s_cnt[myp], 1.0f); }

    // A fragments: xa[d][s_step] holds x[tok=col][d][s_step*4 + 2*half + {0,1}]
    v2f xa[D][4];
    const float* xrow = emb + (size_t)myid * EMB;
    #pragma unroll
    for (int d = 0; d < D; d++) {
        float xn = 0.0f;
        #pragma unroll
        for (int s = 0; s < 4; s++) {
            const v2f a = *(const v2f*)(xrow + d * SUB + s * 4 + half * 2);
            xa[d][s] = a;
            xn += a.x * a.x + a.y * a.y;
        }
        xn += __shfl_xor(xn, 16, 32);
        if (half == 0) xn_lds[wave][col][d] = xn;
    }

    #pragma unroll 1
    for (int kt = 0; kt < 16; kt++) {
        const int k = kt * 16 + col;
        float a0 = 0.f, a1 = 0.f, a2 = 0.f, q0 = 0.f, q1 = 0.f, q2 = 0.f;
        #pragma unroll
        for (int d = 0; d < D; d++) {
            v8f acc = {};
            #pragma unroll
            for (int s = 0; s < 4; s++) {
                const v2f b = *(const v2f*)(cent + (size_t)(d * KC + k) * SUB + s * 4 + half * 2);
                acc = __builtin_amdgcn_wmma_f32_16x16x4_f32(
                    false, xa[d][s], false, b, (short)0, acc, false, false);
            }
            const float cn = cn_lds[d][k];
            #pragma unroll
            for (int v = 0; v < 8; v++) {
                const int row = v + half * 8;
                const float r = 2.0f * acc[v] - xn_lds[wave][row][d] - cn;
                const int p = pt_lds[wave][row];
                const float r2 = r * r;
                a0 += (p == 0) ? r : 0.f;                q0 += (p == 0) ? r2 : 0.f;
                const bool ok1 = (p == 1) && (k < 128);
                a1 += ok1 ? r : 0.f;                     q1 += ok1 ? r2 : 0.f;
                const bool ok2 = (p == 2) && (k < 32);
                a2 += ok2 ? r : 0.f;                     q2 += ok2 ? r2 : 0.f;
            }
        }
        atomicAdd(&s_sum[0][k], a0); atomicAdd(&s_sum2[0][k], q0);
        if (k < 128) { atomicAdd(&s_sum[1][k], a1); atomicAdd(&s_sum2[1][k], q1); }
        if (k <  32) { atomicAdd(&s_sum[2][k], a2); atomicAdd(&s_sum2[2][k], q2); }
    }
    __syncthreads();
    for (int i = tid; i < 3 * KC; i += TPB) {
        atomicAdd(ws + OFF_SUM  + i, (&s_sum[0][0])[i]);
        atomicAdd(ws + OFF_SUM2 + i, (&s_sum2[0][0])[i]);
    }
    if (tid < 3) atomicAdd(ws + OFF_CNT + tid, s_cnt[tid]);
}

// ---------------- Pass 2: mean / inv-std ----------------
__global__ void __launch_bounds__(256) mgqe_finalize_kernel(float* ws) {
    const int i = blockIdx.x * 256 + threadIdx.x;
    if (i >= 3 * KC) return;
    const int p = i >> 8;
    float cnt = ws[OFF_CNT + p] * (float)D;
    cnt = fmaxf(cnt, 1.0f);
    const float mean = ws[OFF_SUM + i] / cnt;
    float var = ws[OFF_SUM2 + i] / cnt - mean * mean;
    var = fmaxf(var, 0.0f);
    ws[OFF_MEAN + i] = mean;
    ws[OFF_ISTD + i] = rsqrtf(var + BN_EPS);
}

// ---------------- Pass 3: recompute resp, normalized argmax, emit output ----------------
__global__ void __launch_bounds__(TPB) mgqe_main_kernel(
    const float* __restrict__ emb, const float* __restrict__ cent,
    const int* __restrict__ ids, const float* __restrict__ ws,
    float* __restrict__ out)
{
    __shared__ float cn_lds[D][KC];
    __shared__ float mean_lds[3][KC];
    __shared__ float istd_lds[3][KC];
    __shared__ float xn_lds[WAVES][16][D];
    __shared__ int   pt_lds[WAVES][16];
    __shared__ int   id_lds[WAVES][16];

    const int tid = threadIdx.x;
    for (int i = tid; i < D * KC; i += TPB) {
        const int d = i >> 8, k = i & 255;
        const float* c = cent + (size_t)(d * KC + k) * SUB;
        float s = 0.0f;
        #pragma unroll
        for (int j = 0; j < SUB; j++) { const float cv = c[j]; s += cv * cv; }
        cn_lds[d][k] = s;
    }
    for (int i = tid; i < 3 * KC; i += TPB) {
        (&mean_lds[0][0])[i] = ws[OFF_MEAN + i];
        (&istd_lds[0][0])[i] = ws[OFF_ISTD + i];
    }
    __syncthreads();

    const int wave = tid >> 5, lane = tid & 31;
    const int col = lane & 15, half = lane >> 4;
    const int tile = blockIdx.x * WAVES + wave;
    const int tok0 = tile * 16;

    const int myid = ids[tok0 + col];
    if (half == 0) { pt_lds[wave][col] = part_of(myid); id_lds[wave][col] = myid; }

    v2f xa[D][4];
    const float* xrow = emb + (size_t)myid * EMB;
    #pragma unroll
    for (int d = 0; d < D; d++) {
        float xn = 0.0f;
        #pragma unroll
        for (int s = 0; s < 4; s++) {
            const v2f a = *(const v2f*)(xrow + d * SUB + s * 4 + half * 2);
            xa[d][s] = a;
            xn += a.x * a.x + a.y * a.y;
        }
        xn += __shfl_xor(xn, 16, 32);
        if (half == 0) xn_lds[wave][col][d] = xn;
    }

    #pragma unroll 1
    for (int d = 0; d < D; d++) {
        float bv[8]; int bi[8];
        #pragma unroll
        for (int v = 0; v < 8; v++) { bv[v] = -__builtin_inff(); bi[v] = 0; }

        #pragma unroll 1
        for (int kt = 0; kt < 16; kt++) {
            const int k = kt * 16 + col;
            v8f acc = {};
            #pragma unroll
            for (int s = 0; s < 4; s++) {
                const v2f b = *(const v2f*)(cent + (size_t)(d * KC + k) * SUB + s * 4 + half * 2);
                acc = __builtin_amdgcn_wmma_f32_16x16x4_f32(
                    false, xa[d][s], false, b, (short)0, acc, false, false);
            }
            const float cn = cn_lds[d][k];
            #pragma unroll
            for (int v = 0; v < 8; v++) {
                const int row = v + half * 8;
                const int p = pt_lds[wave][row];
                const float r  = 2.0f * acc[v] - xn_lds[wave][row][d] - cn;
                const float nv = (r - mean_lds[p][k]) * istd_lds[p][k];
                const bool ok = (k < kp_of(p));
                if (ok && (nv > bv[v])) { bv[v] = nv; bi[v] = k; }
            }
        }
        // argmax across the 16 column lanes of each half (lowest index wins ties)
        #pragma unroll
        for (int v = 0; v < 8; v++) {
            #pragma unroll
            for (int m = 1; m < 16; m <<= 1) {
                const float ov = __shfl_xor(bv[v], m, 32);
                const int   oi = __shfl_xor(bi[v], m, 32);
                if ((ov > bv[v]) || ((ov == bv[v]) && (oi < bi[v]))) { bv[v] = ov; bi[v] = oi; }
            }
        }
        // emit: lane writes element s=col of rows v(+8); out = (q - x) + x
        #pragma unroll
        for (int v = 0; v < 8; v++) {
            const int row  = v + half * 8;
            const int code = bi[v];
            const int rid  = id_lds[wave][row];
            const float xv = emb[(size_t)rid * EMB + d * SUB + col];
            const float qv = cent[(size_t)(d * KC + code) * SUB + col];
            out[(size_t)(tok0 + row) * EMB + d * SUB + col] = (qv - xv) + xv;
        }
    }
}

extern "C" void kernel_launch(void* const* d_in, const int* in_sizes, int n_in,
                              void* d_out, int out_size, void* d_ws, size_t ws_size,
                              hipStream_t stream) {
    const float* emb  = (const float*)d_in[0];
    const float* cent = (const float*)d_in[1];
    const int*   ids  = (const int*)d_in[2];
    float* out = (float*)d_out;
    float* ws  = (float*)d_ws;

    init_ws_kernel<<<1, 256, 0, stream>>>(ws);
    mgqe_stats_kernel<<<BLOCKS, TPB, 0, stream>>>(emb, cent, ids, ws);
    mgqe_finalize_kernel<<<3, 256, 0, stream>>>(ws);
    mgqe_main_kernel<<<BLOCKS, TPB, 0, stream>>>(emb, cent, ids, ws, out);
}